// SpatialAtt_50019189129335
// MI455X (gfx1250) — compile-verified
//
#include <hip/hip_runtime.h>
#include <math.h>

#define N_VOX 4096
#define C_DIM 128

typedef float v2f __attribute__((ext_vector_type(2)));
typedef float v8f __attribute__((ext_vector_type(8)));

// ---- Kernel 1: att[b,n] = sum_c x[b,c,n]*w[c] + map_b, plus per-block min/max ----
__global__ void __launch_bounds__(256)
k_att(const float* __restrict__ x, const float* __restrict__ map_w,
      const float* __restrict__ map_b, float* __restrict__ att,
      float* __restrict__ pmin, float* __restrict__ pmax) {
    const int idx = blockIdx.x * 256 + threadIdx.x;   // 0..8191
    const int b = idx >> 12;
    const int n = idx & (N_VOX - 1);
    const float* xp = x + ((size_t)b * C_DIM) * N_VOX + n;
    float acc = 0.f;
#pragma unroll 8
    for (int c = 0; c < C_DIM; ++c)
        acc = fmaf(xp[(size_t)c * N_VOX], map_w[c], acc);   // w[c] uniform -> s_load
    const float a = acc + map_b[0];
    att[idx] = a;

    __shared__ float smin[256], smax[256];
    smin[threadIdx.x] = a; smax[threadIdx.x] = a;
    __syncthreads();
    for (int off = 128; off > 0; off >>= 1) {
        if (threadIdx.x < off) {
            smin[threadIdx.x] = fminf(smin[threadIdx.x], smin[threadIdx.x + off]);
            smax[threadIdx.x] = fmaxf(smax[threadIdx.x], smax[threadIdx.x + off]);
        }
        __syncthreads();
    }
    if (threadIdx.x == 0) { pmin[blockIdx.x] = smin[0]; pmax[blockIdx.x] = smax[0]; }
}

// ---- Kernel 1b: reduce 16 block-partials per batch (lanes 0-15 = b0, 16-31 = b1) ----
__global__ void __launch_bounds__(32)
k_minmax(const float* __restrict__ pmin, const float* __restrict__ pmax,
         float* __restrict__ attmin, float* __restrict__ attmax) {
    const int t = threadIdx.x, b = t >> 4, l = t & 15;
    float vmin = pmin[b * 16 + l], vmax = pmax[b * 16 + l];
#pragma unroll
    for (int off = 8; off > 0; off >>= 1) {
        vmin = fminf(vmin, __shfl_xor(vmin, off, 32));
        vmax = fmaxf(vmax, __shfl_xor(vmax, off, 32));
    }
    if (l == 0) { attmin[b] = vmin; attmax[b] = vmax; }
}

// ---- Kernel 2: per 16-row tile, one wave: stable softmax row stats via WMMA ----
// acc D[i][0] = S_i = sum_j exp(q_i*k_j - m_i),  D[i][1] = T_i = sum_j exp(..)*att_j
__global__ void __launch_bounds__(32)
k_softmax(const float* __restrict__ att, const float* __restrict__ attminA,
          const float* __restrict__ attmaxA,
          const float* __restrict__ qws, const float* __restrict__ qbs,
          const float* __restrict__ kws, const float* __restrict__ kbs,
          float* __restrict__ g) {
    const int lane = threadIdx.x;
    const int tile = blockIdx.x & 255;
    const int b    = blockIdx.x >> 8;
    const float q_w = qws[0], q_b = qbs[0], k_w = kws[0], k_b = kbs[0];

    __shared__ __align__(16) float s_att[N_VOX];
    const float* attb = att + b * N_VOX;
    for (int t = lane; t < N_VOX; t += 32) s_att[t] = attb[t];
    __syncthreads();

    // k_j = att_j*k_w + k_b is monotone in att_j -> k range from att range
    const float amax = attmaxA[b], amin = attminA[b];
    const float kmax = (k_w >= 0.f) ? fmaf(amax, k_w, k_b) : fmaf(amin, k_w, k_b);
    const float kmin = (k_w >= 0.f) ? fmaf(amin, k_w, k_b) : fmaf(amax, k_w, k_b);

    // A-matrix 16x4 f32 layout: lanes 0-15 row M=lane (K=0,1), lanes 16-31 row M=lane-16 (K=2,3)
    const int   iRow  = lane & 15;
    const float att_i = s_att[tile * 16 + iRow];
    const float q_i   = fmaf(att_i, q_w, q_b);
    const float m_i   = (q_i >= 0.f) ? q_i * kmax : q_i * kmin;   // max_j q_i*k_j

    const bool hi   = lane >= 16;          // K=2,3 half
    const bool isC0 = (iRow == 0);         // B column 0 -> ones  (S)
    const bool isC1 = (iRow == 1);         // B column 1 -> att_j (T)

    v8f acc0 = {}; v8f acc1 = {};          // dual accumulators: break D->C chain
    for (int j0 = 0; j0 < N_VOX; j0 += 8) {
        const float4 ajA = *(const float4*)&s_att[j0];
        const float4 ajB = *(const float4*)&s_att[j0 + 4];
        // ---- chunk A (j0..j0+3) ----
        {
            const float a0 = hi ? ajA.z : ajA.x;
            const float a1 = hi ? ajA.w : ajA.y;
            const float e0 = __expf(fmaf(q_i, fmaf(a0, k_w, k_b), -m_i));
            const float e1 = __expf(fmaf(q_i, fmaf(a1, k_w, k_b), -m_i));
            v2f A; A.x = e0; A.y = e1;
            v2f B; B.x = isC0 ? 1.f : (isC1 ? a0 : 0.f);
                   B.y = isC0 ? 1.f : (isC1 ? a1 : 0.f);
            acc0 = __builtin_amdgcn_wmma_f32_16x16x4_f32(false, A, false, B,
                                                         (short)0, acc0, false, false);
        }
        // ---- chunk B (j0+4..j0+7) ----
        {
            const float a0 = hi ? ajB.z : ajB.x;
            const float a1 = hi ? ajB.w : ajB.y;
            const float e0 = __expf(fmaf(q_i, fmaf(a0, k_w, k_b), -m_i));
            const float e1 = __expf(fmaf(q_i, fmaf(a1, k_w, k_b), -m_i));
            v2f A; A.x = e0; A.y = e1;
            v2f B; B.x = isC0 ? 1.f : (isC1 ? a0 : 0.f);
                   B.y = isC0 ? 1.f : (isC1 ? a1 : 0.f);
            acc1 = __builtin_amdgcn_wmma_f32_16x16x4_f32(false, A, false, B,
                                                         (short)0, acc1, false, false);
        }
    }

    // D layout: VGPR r -> rows M=r (lanes 0-15, N=lane) and M=r+8 (lanes 16-31, N=lane-16)
    float* gout = g + b * N_VOX + tile * 16;
#pragma unroll
    for (int r = 0; r < 8; ++r) {
        const float v = acc0[r] + acc1[r];
        const float S0 = __shfl(v, 0, 32);
        const float T0 = __shfl(v, 1, 32);
        const float S1 = __shfl(v, 16, 32);
        const float T1 = __shfl(v, 17, 32);
        if (lane == 0) {
            const float o = T0 / S0;
            gout[r] = o / (1.f + fabsf(o));          // soft_sign
        } else if (lane == 1) {
            const float o = T1 / S1;
            gout[r + 8] = o / (1.f + fabsf(o));
        }
    }
}

// ---- Kernel 3: y[b,c,n] = g[b,n] * x[b,c,n]  (float4, fully coalesced) ----
__global__ void __launch_bounds__(256)
k_scale(const float* __restrict__ x, const float* __restrict__ g,
        float* __restrict__ y) {
    const int idx = blockIdx.x * 256 + threadIdx.x;  // 0..262143 (float4 units)
    const int e = idx << 2;
    const int b = e >> 19;                           // 128*4096 = 2^19 elems per batch
    const int n = e & (N_VOX - 1);
    const float4 xv = *(const float4*)(x + e);
    const float4 gv = *(const float4*)(g + b * N_VOX + n);
    float4 o;
    o.x = xv.x * gv.x; o.y = xv.y * gv.y; o.z = xv.z * gv.z; o.w = xv.w * gv.w;
    *(float4*)(y + e) = o;
}

extern "C" void kernel_launch(void* const* d_in, const int* in_sizes, int n_in,
                              void* d_out, int out_size, void* d_ws, size_t ws_size,
                              hipStream_t stream) {
    const float* x     = (const float*)d_in[0];
    const float* map_w = (const float*)d_in[1];
    const float* map_b = (const float*)d_in[2];
    const float* q_w   = (const float*)d_in[3];
    const float* q_b   = (const float*)d_in[4];
    const float* k_w   = (const float*)d_in[5];
    const float* k_b   = (const float*)d_in[6];

    float* ws     = (float*)d_ws;
    float* att    = ws;            // 2*4096
    float* g      = ws + 8192;     // 2*4096
    float* pmin   = ws + 16384;    // 32
    float* pmax   = ws + 16416;    // 32
    float* attmin = ws + 16448;    // 2
    float* attmax = ws + 16450;    // 2

    k_att    <<<32,   256, 0, stream>>>(x, map_w, map_b, att, pmin, pmax);
    k_minmax <<<1,    32,  0, stream>>>(pmin, pmax, attmin, attmax);
    k_softmax<<<512,  32,  0, stream>>>(att, attmin, attmax, q_w, q_b, k_w, k_b, g);
    k_scale  <<<1024, 256, 0, stream>>>(x, g, (float*)d_out);
}